// FixedPointIterationGap4_88175678587250
// MI455X (gfx1250) — compile-verified
//
#include <hip/hip_runtime.h>
#include <hip/hip_bf16.h>

// ---------------------------------------------------------------------------
// Smoothed chain DP (forward-backward logsumexp scan) for MI455X / gfx1250.
// One wave32 per chain; C=21 states live in lanes 0..20.
// Horizontal chains: pairwise tile [441 rows x T steps] is TDM-friendly
// (constant row stride, contiguous t) -> tensor_load_to_lds double buffering.
// Vertical chains: contiguous axis is the chain index -> direct loads + L2.
// ---------------------------------------------------------------------------

#define CDIM 21
#define CC   441        // 21*21
#define TT   8          // TDM tile width (time steps per buffer)
#define MAXL 96

typedef unsigned int u32x4 __attribute__((ext_vector_type(4)));
typedef int          i32x8 __attribute__((ext_vector_type(8)));
typedef int          i32x4 __attribute__((ext_vector_type(4)));

#if defined(__HIP_DEVICE_COMPILE__) && __has_builtin(__builtin_amdgcn_tensor_load_to_lds)
#define HAVE_TDM 1
#else
#define HAVE_TDM 0
#endif

#if defined(__HIP_DEVICE_COMPILE__) && __has_builtin(__builtin_amdgcn_s_wait_tensorcnt)
#define WAIT_TENSOR(n) __builtin_amdgcn_s_wait_tensorcnt(n)
#else
#define WAIT_TENSOR(n) ((void)0)
#endif

// Issue one TDM 2D-tile load: nrows rows of nt contiguous 4-byte elements,
// rows separated by row_stride_elems, packed densely into LDS at lds_byte_off.
__device__ __forceinline__ void tdm_load_2d(unsigned lds_byte_off,
                                            const float* gptr,
                                            int nt, int nrows,
                                            long row_stride_elems) {
#if HAVE_TDM
  unsigned long long ga = (unsigned long long)(size_t)gptr;
  unsigned long long rs = (unsigned long long)row_stride_elems;
  u32x4 g0; i32x8 g1; i32x4 g2; i32x4 g3; i32x8 g4;
  // D# group 0: count=1, lds_addr, 57-bit global addr, type=2
  g0[0] = 1u;
  g0[1] = lds_byte_off;
  g0[2] = (unsigned)(ga & 0xffffffffull);
  g0[3] = (unsigned)((ga >> 32) & 0x01ffffffull) | (2u << 30);
  // D# group 1: data_size=4B; tensor_dim0=nt, tensor_dim1=nrows;
  // tile_dim0=nt, tile_dim1=nrows, tile_dim2=0; dim0 stride = row stride.
  g1[0] = 0x20000;                                                  // data_size=2 (4B)
  g1[1] = (int)(((unsigned)nt & 0xffffu) << 16);                    // tensor_dim0[15:0]
  g1[2] = (int)((((unsigned)nt >> 16) & 0xffffu) |
                (((unsigned)nrows & 0xffffu) << 16));               // td0 hi | td1 lo
  g1[3] = (int)((((unsigned)nrows >> 16) & 0xffffu) |
                (((unsigned)nt & 0xffffu) << 16));                  // td1 hi | tile_dim0
  g1[4] = (int)((unsigned)nrows & 0xffffu);                         // tile_dim1 | tile_dim2=0
  g1[5] = (int)(unsigned)(rs & 0xffffffffull);                      // dim0 stride lo32
  g1[6] = (int)(unsigned)((rs >> 32) & 0xffffull);                  // dim0 stride hi16
  g1[7] = 0;
  g2[0] = g2[1] = g2[2] = g2[3] = 0;
  g3[0] = g3[1] = g3[2] = g3[3] = 0;
  g4[0] = g4[1] = g4[2] = g4[3] = 0;
  g4[4] = g4[5] = g4[6] = g4[7] = 0;
  __builtin_amdgcn_tensor_load_to_lds(g0, g1, g2, g3, g4, 0);
#else
  (void)lds_byte_off; (void)gptr; (void)nt; (void)nrows; (void)row_stride_elems;
#endif
}

__device__ __forceinline__ int imin(int a, int b) { return a < b ? a : b; }

// One wave32 per chain. HORIZ: chains run along W; else along H.
// u layout [B,C,Hg,Wg]; p layout [B,C,C,Hg,Wg-1] (HORIZ) or [B,C,C,Hg-1,Wg].
// blockIdx.y selects the grid (for the stacked half/third inputs).
template <bool HORIZ, bool USE_TDM>
__global__ __launch_bounds__(32) void dp_kernel(
    const float* __restrict__ u, const float* __restrict__ p,
    float* __restrict__ mout, long u_gstride, long p_gstride,
    int B, int Hg, int Wg, const float* __restrict__ gammap) {
  __shared__ float alpha[MAXL][CDIM];
  __shared__ float cur[CDIM];   // running alpha / beta+unary vector
  __shared__ float bn[CDIM];
  __shared__ float red[CDIM];
  __shared__ float pbuf[USE_TDM ? 2 * CC * TT : 1];

  const int lane  = threadIdx.x;
  const int gidx  = blockIdx.y;
  const int q     = blockIdx.x;
  const int nline = HORIZ ? Hg : Wg;
  const int L     = HORIZ ? Wg : Hg;
  const int Lp    = L - 1;
  const int bb    = q / nline;
  const int line  = q - bb * nline;

  const int  su_c   = Hg * Wg;
  const int  su_t   = HORIZ ? 1 : Wg;
  const long sp_row = HORIZ ? (long)Hg * (Wg - 1) : (long)(Hg - 1) * Wg;
  const int  sp_t   = HORIZ ? 1 : Wg;

  const long ubase = (long)gidx * u_gstride + (long)bb * CDIM * su_c +
                     (HORIZ ? (long)line * Wg : (long)line);
  const long pbase = (long)gidx * p_gstride + (long)bb * CC * sp_row +
                     (HORIZ ? (long)line * (Wg - 1) : (long)line);

  const float g    = fmaxf(gammap[0], 0.01f);
  const float inv  = 1.0f / g;
  const bool  act  = lane < CDIM;

  // ---------------- forward: alpha ----------------
  if (act) {
    float a0 = u[ubase + (long)lane * su_c];
    alpha[0][lane] = a0;
    cur[lane] = a0;
  }
  __syncthreads();

  if constexpr (USE_TDM) {
    const int ntiles = (Lp + TT - 1) / TT;
    tdm_load_2d((unsigned)(size_t)&pbuf[0], p + pbase, imin(TT, Lp), CC, sp_row);
    for (int tb = 0; tb < ntiles; ++tb) {
      const int nt = imin(TT, Lp - tb * TT);
      if (tb + 1 < ntiles) {
        tdm_load_2d((unsigned)(size_t)&pbuf[((tb + 1) & 1) * CC * TT],
                    p + pbase + (long)(tb + 1) * TT,
                    imin(TT, Lp - (tb + 1) * TT), CC, sp_row);
        WAIT_TENSOR(1);
      } else {
        WAIT_TENSOR(0);
      }
      const float* tile = &pbuf[(tb & 1) * CC * TT];
      for (int tt = 0; tt < nt; ++tt) {
        const int t = tb * TT + tt + 1;
        float anew = 0.0f;
        if (act) {
          float pv[CDIM];
#pragma unroll
          for (int i = 0; i < CDIM; ++i)
            pv[i] = tile[(i * CDIM + lane) * nt + tt] + cur[i];
          float mx = pv[0];
#pragma unroll
          for (int i = 1; i < CDIM; ++i) mx = fmaxf(mx, pv[i]);
          float s = 0.0f;
#pragma unroll
          for (int i = 0; i < CDIM; ++i) s += __expf((pv[i] - mx) * inv);
          anew = u[ubase + (long)lane * su_c + (long)t * su_t] + mx + g * __logf(s);
        }
        __syncthreads();
        if (act) { alpha[t][lane] = anew; cur[lane] = anew; }
        __syncthreads();
      }
    }
  } else {
    for (int t = 1; t < L; ++t) {
      const long pb = pbase + (long)(t - 1) * sp_t;
      float anew = 0.0f;
      if (act) {
        float pv[CDIM];
#pragma unroll
        for (int i = 0; i < CDIM; ++i)
          pv[i] = p[pb + (long)(i * CDIM + lane) * sp_row] + cur[i];
        float mx = pv[0];
#pragma unroll
        for (int i = 1; i < CDIM; ++i) mx = fmaxf(mx, pv[i]);
        float s = 0.0f;
#pragma unroll
        for (int i = 0; i < CDIM; ++i) s += __expf((pv[i] - mx) * inv);
        anew = u[ubase + (long)lane * su_c + (long)t * su_t] + mx + g * __logf(s);
      }
      __syncthreads();
      if (act) { alpha[t][lane] = anew; cur[lane] = anew; }
      __syncthreads();
    }
  }

  // ---------------- marginal at t = L-1, init bn ----------------
  if (act) red[lane] = alpha[L - 1][lane] * inv;
  __syncthreads();
  if (act) {
    float mx = red[0];
#pragma unroll
    for (int i = 1; i < CDIM; ++i) mx = fmaxf(mx, red[i]);
    float s = 0.0f;
#pragma unroll
    for (int i = 0; i < CDIM; ++i) s += __expf(red[i] - mx);
    const long uo = ubase + (long)lane * su_c + (long)(L - 1) * su_t;
    mout[uo] = __expf(red[lane] - mx) / s;
    bn[lane] = u[uo];  // beta_{L-1} = 0
  }
  __syncthreads();

  // ---------------- backward: beta + marginals ----------------
  if constexpr (USE_TDM) {
    const int ntiles = (Lp + TT - 1) / TT;
    int tb0 = ntiles - 1;
    tdm_load_2d((unsigned)(size_t)&pbuf[(tb0 & 1) * CC * TT],
                p + pbase + (long)tb0 * TT, imin(TT, Lp - tb0 * TT), CC, sp_row);
    for (int tb = tb0; tb >= 0; --tb) {
      const int nt = imin(TT, Lp - tb * TT);
      if (tb - 1 >= 0) {
        tdm_load_2d((unsigned)(size_t)&pbuf[((tb - 1) & 1) * CC * TT],
                    p + pbase + (long)(tb - 1) * TT, TT, CC, sp_row);
        WAIT_TENSOR(1);
      } else {
        WAIT_TENSOR(0);
      }
      const float* tile = &pbuf[(tb & 1) * CC * TT];
      for (int tt = nt - 1; tt >= 0; --tt) {
        const int t = tb * TT + tt;  // pairwise index t (between t and t+1)
        float bnew = 0.0f, x = 0.0f;
        if (act) {
          float pv[CDIM];
#pragma unroll
          for (int j = 0; j < CDIM; ++j)
            pv[j] = tile[(lane * CDIM + j) * nt + tt] + bn[j];
          float mx = pv[0];
#pragma unroll
          for (int j = 1; j < CDIM; ++j) mx = fmaxf(mx, pv[j]);
          float s = 0.0f;
#pragma unroll
          for (int j = 0; j < CDIM; ++j) s += __expf((pv[j] - mx) * inv);
          bnew = mx + g * __logf(s);
          x = (alpha[t][lane] + bnew) * inv;
          red[lane] = x;
        }
        __syncthreads();
        if (act) {
          float mx2 = red[0];
#pragma unroll
          for (int i = 1; i < CDIM; ++i) mx2 = fmaxf(mx2, red[i]);
          float s2 = 0.0f;
#pragma unroll
          for (int i = 0; i < CDIM; ++i) s2 += __expf(red[i] - mx2);
          const long uo = ubase + (long)lane * su_c + (long)t * su_t;
          mout[uo] = __expf(x - mx2) / s2;
          bn[lane] = bnew + u[uo];
        }
        __syncthreads();
      }
    }
  } else {
    for (int t = L - 2; t >= 0; --t) {
      const long pb = pbase + (long)t * sp_t;
      float bnew = 0.0f, x = 0.0f;
      if (act) {
        float pv[CDIM];
#pragma unroll
        for (int j = 0; j < CDIM; ++j)
          pv[j] = p[pb + (long)(lane * CDIM + j) * sp_row] + bn[j];
        float mx = pv[0];
#pragma unroll
        for (int j = 1; j < CDIM; ++j) mx = fmaxf(mx, pv[j]);
        float s = 0.0f;
#pragma unroll
        for (int j = 0; j < CDIM; ++j) s += __expf((pv[j] - mx) * inv);
        bnew = mx + g * __logf(s);
        x = (alpha[t][lane] + bnew) * inv;
        red[lane] = x;
      }
      __syncthreads();
      if (act) {
        float mx2 = red[0];
#pragma unroll
        for (int i = 1; i < CDIM; ++i) mx2 = fmaxf(mx2, red[i]);
        float s2 = 0.0f;
#pragma unroll
        for (int i = 0; i < CDIM; ++i) s2 += __expf(red[i] - mx2);
        const long uo = ubase + (long)lane * su_c + (long)t * su_t;
        mout[uo] = __expf(x - mx2) / s2;
        bn[lane] = bnew + u[uo];
      }
      __syncthreads();
    }
  }
}

// ---------------------------------------------------------------------------
// Combine: scatter-add half/third marginals into the full grid, compute avg,
// unary updates and the accumulated mh/mv outputs. Full<->half<->third pixel
// mappings are bijections, so one thread per full-grid element writes all 8
// outputs exactly once.
// ---------------------------------------------------------------------------
__global__ __launch_bounds__(256) void combine_kernel(
    const float* __restrict__ uh0, const float* __restrict__ uv0,
    const float* __restrict__ uhh, const float* __restrict__ uvh,
    const float* __restrict__ uht, const float* __restrict__ uvt,
    const float* __restrict__ mh0, const float* __restrict__ mv0,
    const float* __restrict__ mhh, const float* __restrict__ mvh,
    const float* __restrict__ mht, const float* __restrict__ mvt,
    float* __restrict__ out) {
  const long G = 4L * 21 * 96 * 96;  // 774144
  long idx = (long)blockIdx.x * blockDim.x + threadIdx.x;
  if (idx >= G) return;
  int x = (int)(idx % 96);
  long tmp = idx / 96;
  int y = (int)(tmp % 96);
  tmp /= 96;
  int c = (int)(tmp % 21);
  int bb = (int)(tmp / 21);

  const int k2 = (y & 1) * 2 + (x & 1);
  const long off2 =
      ((((long)k2 * 4 + bb) * 21 + c) * 48 + (y >> 1)) * 48 + (x >> 1);
  const int y3 = y / 3, x3 = x / 3;
  const int k3 = (y - y3 * 3) * 3 + (x - x3 * 3);
  const long off3 = ((((long)k3 * 4 + bb) * 21 + c) * 32 + y3) * 32 + x3;

  const float h0 = mh0[idx], v0 = mv0[idx];
  const float h2 = mhh[off2], v2 = mvh[off2];
  const float h3 = mht[off3], v3 = mvt[off3];
  const float mh = h0 + h2 + h3;
  const float mv = v0 + v2 + v3;
  const float avg = (mh + mv) * (1.0f / 6.0f);
  const float invD = 1.0f / 96.0f;  // W == H == 96 used for every scale

  out[0 * G + idx]  = uh0[idx]  - (h0 - avg) * invD;
  out[1 * G + off2] = uhh[off2] - (h2 - avg) * invD;
  out[2 * G + off3] = uht[off3] - (h3 - avg) * invD;
  out[3 * G + idx]  = uv0[idx]  - (v0 - avg) * invD;
  out[4 * G + off2] = uvh[off2] - (v2 - avg) * invD;
  out[5 * G + off3] = uvt[off3] - (v3 - avg) * invD;
  out[6 * G + idx]  = mh;
  out[7 * G + idx]  = mv;
}

extern "C" void kernel_launch(void* const* d_in, const int* in_sizes, int n_in,
                              void* d_out, int out_size, void* d_ws,
                              size_t ws_size, hipStream_t stream) {
  (void)in_sizes; (void)n_in; (void)out_size; (void)ws_size;
  const float* uh0 = (const float*)d_in[0];
  const float* uv0 = (const float*)d_in[1];
  const float* ph0 = (const float*)d_in[2];
  const float* pv0 = (const float*)d_in[3];
  const float* uhh = (const float*)d_in[4];
  const float* uvh = (const float*)d_in[5];
  const float* phh = (const float*)d_in[6];
  const float* pvh = (const float*)d_in[7];
  const float* uht = (const float*)d_in[8];
  const float* uvt = (const float*)d_in[9];
  const float* pht = (const float*)d_in[10];
  const float* pvt = (const float*)d_in[11];
  const float* gamma = (const float*)d_in[12];

  const long G = 4L * 21 * 96 * 96;  // 774144 elements per full-grid tensor
  float* ws = (float*)d_ws;          // needs 6*G floats (~18.6 MB)
  float* mh0 = ws + 0 * G;
  float* mv0 = ws + 1 * G;
  float* mhh = ws + 2 * G;
  float* mvh = ws + 3 * G;
  float* mht = ws + 4 * G;
  float* mvt = ws + 5 * G;

  const dim3 blk(32);
  // full scale: 384 chains each direction
  dp_kernel<true, true><<<dim3(4 * 96, 1), blk, 0, stream>>>(
      uh0, ph0, mh0, 0L, 0L, 4, 96, 96, gamma);
  dp_kernel<false, false><<<dim3(4 * 96, 1), blk, 0, stream>>>(
      uv0, pv0, mv0, 0L, 0L, 4, 96, 96, gamma);
  // half scale: 4 grids x 192 chains
  dp_kernel<true, true><<<dim3(4 * 48, 4), blk, 0, stream>>>(
      uhh, phh, mhh, 193536L, 3979584L, 4, 48, 48, gamma);
  dp_kernel<false, false><<<dim3(4 * 48, 4), blk, 0, stream>>>(
      uvh, pvh, mvh, 193536L, 3979584L, 4, 48, 48, gamma);
  // third scale: 9 grids x 128 chains
  dp_kernel<true, true><<<dim3(4 * 32, 9), blk, 0, stream>>>(
      uht, pht, mht, 86016L, 1749888L, 4, 32, 32, gamma);
  dp_kernel<false, false><<<dim3(4 * 32, 9), blk, 0, stream>>>(
      uvt, pvt, mvt, 86016L, 1749888L, 4, 32, 32, gamma);

  const int nthreads = 256;
  const int nblocks = (int)((G + nthreads - 1) / nthreads);
  combine_kernel<<<nblocks, nthreads, 0, stream>>>(
      uh0, uv0, uhh, uvh, uht, uvt, mh0, mv0, mhh, mvh, mht, mvt,
      (float*)d_out);
}